// SequenceModel_36129264894522
// MI455X (gfx1250) — compile-verified
//
#include <hip/hip_runtime.h>
#include <math.h>

// Problem constants (match reference)
#define Bn   256
#define Tn   32
#define DQn  128
#define Kn   8
#define Sn   4
#define CTXn 16
#define Dn   (2*DQn + 1)   // 257
#define EVOn 8
#define BTILE 16
#define NWG  (Bn / BTILE)  // 16
#define STRIDE 130         // padded LDS row stride (dwords): 8B-aligned, conflict-free

typedef float v2f __attribute__((ext_vector_type(2)));
typedef float v8f __attribute__((ext_vector_type(8)));

// d_ws accumulator layout (raw sums over all (b,t)):
// [0] sumF  [1] sumCorrect  [2] sumCE  [3..10] usage[8]
// [11..12] tsum[2]  [13..14] tcnt[2]  [15..46] scw[4][8]  [47..50] scnt[4]
#define NACC 51

__global__ void seqmodel_init(float* __restrict__ ws) {
  int i = threadIdx.x;
  if (i < NACC) ws[i] = 0.0f;
}

__global__ __launch_bounds__(256)
void seqmodel_main(const int* __restrict__ inputs, const int* __restrict__ targets,
                   const int* __restrict__ task_ids, const float* __restrict__ init_state,
                   const float* __restrict__ bit_embed, const float* __restrict__ task_u_embed,
                   const float* __restrict__ rce,   // router_ctx_embed (2,CTX)
                   const float* __restrict__ rw,    // readout_weight (K,S,D)
                   const float* __restrict__ rb,    // readout_bias (K,S)
                   const float* __restrict__ Wq,    // (DQ,K)
                   const float* __restrict__ Wc,    // (CTX,K)
                   const float* __restrict__ rbias, // router_b (K)
                   const float* __restrict__ Amat,  // (K,DQ,DQ)
                   float* __restrict__ ws_accum)
{
  __shared__ float q [BTILE][STRIDE];
  __shared__ float p [BTILE][STRIDE];
  __shared__ float tq[BTILE][STRIDE];
  __shared__ float uin[BTILE][STRIDE];
  __shared__ float wq_s[DQn][Kn];      // LDS-cached router_Wq (4 KB)
  __shared__ float s_col[BTILE];
  __shared__ float wlast[BTILE][Kn];   // chart_w (last evo step)
  __shared__ float w2[BTILE][Kn];      // chart_w_next
  __shared__ float ctxb[BTILE][Kn];    // ctx@Wc + b (per timestep)
  __shared__ float logits4[BTILE][4];
  __shared__ float wn_l[BTILE][2];
  __shared__ int   idx_l[BTILE][2];
  __shared__ int   bit_t[BTILE], tgt_t[BTILE], prev_t[BTILE], task_t[BTILE];
  __shared__ float accum[NACC];

  const int tid  = threadIdx.x;
  const int lane = tid & 31;
  const int wave = tid >> 5;      // 0..7
  const int half = lane >> 4;     // lane group (K split for WMMA frags)
  const int lm   = lane & 15;     // row/col within 16
  const int row0 = blockIdx.x * BTILE;

  const float DT = 0.2f, DAMP = 0.1f, GAMMA = 10.0f, EPS = 1e-8f;
  const float pk = 1.0f - DT * DAMP;

  for (int i = tid; i < NACC; i += 256) accum[i] = 0.0f;

  // Cache router_Wq in LDS
  for (int i = tid; i < DQn * Kn; i += 256) wq_s[i >> 3][i & 7] = Wq[i];

  // Load initial state tile: q | p | s
  for (int i = tid; i < BTILE * DQn; i += 256) {
    int r = i / DQn, c = i % DQn;
    q[r][c] = init_state[(size_t)(row0 + r) * Dn + c];
    p[r][c] = init_state[(size_t)(row0 + r) * Dn + DQn + c];
  }
  if (tid < BTILE) s_col[tid] = init_state[(size_t)(row0 + tid) * Dn + 2 * DQn];
  __syncthreads();

  for (int t = 0; t < Tn; ++t) {
    // ---- per-timestep inputs ----
    if (tid < BTILE) {
      int rg = row0 + tid;
      bit_t[tid]  = inputs[rg * Tn + t];
      tgt_t[tid]  = targets[rg * Tn + t];
      prev_t[tid] = (t == 0) ? 0 : targets[rg * Tn + t - 1];
      task_t[tid] = task_ids[rg * Tn + t];
    }
    __syncthreads();
    for (int i = tid; i < BTILE * DQn; i += 256) {
      int r = i / DQn, c = i % DQn;
      uin[r][c] = tanhf(bit_embed[bit_t[r] * DQn + c]) +
                  tanhf(task_u_embed[task_t[r] * DQn + c]);
    }
    if (tid < BTILE * Kn) {
      int r = tid >> 3, k = tid & 7;
      float acc = rbias[k];
      for (int c = 0; c < CTXn; ++c)
        acc += tanhf(rce[task_t[r] * CTXn + c]) * Wc[c * Kn + k];
      ctxb[r][k] = acc;
    }
    __syncthreads();

    // ---- EVO inner scan ----
    for (int e = 0; e < EVOn; ++e) {
      // tq = tanh(q)
      for (int i = tid; i < BTILE * DQn; i += 256) {
        int r = i / DQn, c = i % DQn;
        tq[r][c] = tanhf(q[r][c]);
      }
      __syncthreads();
      // router logits = q@Wq + ctxb  (TAU == 1)
      if (tid < BTILE * Kn) {
        int r = tid >> 3, k = tid & 7;
        float acc = ctxb[r][k];
        for (int j = 0; j < DQn; ++j) acc += q[r][j] * wq_s[j][k];
        wlast[r][k] = acc;
      }
      __syncthreads();
      // softmax over K per row
      if (tid < BTILE) {
        float m = wlast[tid][0];
        for (int k = 1; k < Kn; ++k) m = fmaxf(m, wlast[tid][k]);
        float s = 0.0f, ex[Kn];
        for (int k = 0; k < Kn; ++k) { ex[k] = expf(wlast[tid][k] - m); s += ex[k]; }
        for (int k = 0; k < Kn; ++k) wlast[tid][k] = ex[k] / s;
      }
      __syncthreads();

      // ---- force = sum_k w[b,k] * (tq @ A_k^T), WMMA f32 16x16x4 ----
      // wave owns output columns [n0, n0+16); exclusive p/q cell ownership.
      // 4 independent accumulator chains over the j reduction for WMMA ILP.
      {
        const int n0   = wave * 16;
        const int icol = n0 + lm;           // output i index (N = lm for all lanes)
        v8f facc;
        #pragma unroll
        for (int u = 0; u < 8; ++u) facc[u] = 0.0f;
        for (int k = 0; k < Kn; ++k) {
          v8f cc[4];
          #pragma unroll
          for (int h4 = 0; h4 < 4; ++h4)
            #pragma unroll
            for (int u = 0; u < 8; ++u) cc[h4][u] = 0.0f;
          // B[j][i] = A[k][i][j]; this lane covers K-rows j0+2*half, j0+2*half+1
          const float* Abase = Amat + ((size_t)(k * DQn + icol)) * DQn + 2 * half;
#if __has_builtin(__builtin_amdgcn_wmma_f32_16x16x4_f32)
          #pragma unroll
          for (int jc = 0; jc < 4; ++jc) {
            #pragma unroll
            for (int js = 0; js < 32; js += 4) {
              const int j0 = jc * 32 + js;
              v2f a, b;
              a.x = tq[lm][j0 + 2 * half];        // A-frag: M=lm, K=j0+2h / +1
              a.y = tq[lm][j0 + 2 * half + 1];
              b.x = Abase[j0];                    // B-frag: K=j0+2h / +1, N=lm
              b.y = Abase[j0 + 1];
              cc[jc] = __builtin_amdgcn_wmma_f32_16x16x4_f32(
                           false, a, false, b, (short)0, cc[jc], false, false);
            }
          }
          #pragma unroll
          for (int u = 0; u < 8; ++u) {
            int m = u + 8 * half;               // C/D layout: lanes16-31 -> M+8
            facc[u] += wlast[m][k] * (cc[0][u] + cc[1][u] + cc[2][u] + cc[3][u]);
          }
#else
          // scalar fallback (same ownership mapping)
          for (int u = 0; u < 8; ++u) {
            int m = u + 8 * half;
            float dk = 0.0f;
            const float* Ap = Amat + ((size_t)(k * DQn + icol)) * DQn;
            for (int j = 0; j < DQn; ++j) dk += Ap[j] * tq[m][j];
            facc[u] += wlast[m][k] * dk;
          }
          (void)cc; (void)Abase;
#endif
        }
        // p,q update on exclusively-owned cells
        #pragma unroll
        for (int u = 0; u < 8; ++u) {
          int m = u + 8 * half;
          float pn = pk * p[m][icol] + DT * (facc[u] + uin[m][icol]);
          p[m][icol] = pn;
          q[m][icol] = q[m][icol] + DT * tanhf(pn);
        }
      }
      __syncthreads();
    } // evo

    // ---- chart_w_next = router(q_final) ----
    if (tid < BTILE * Kn) {
      int r = tid >> 3, k = tid & 7;
      float acc = ctxb[r][k];
      for (int j = 0; j < DQn; ++j) acc += q[r][j] * wq_s[j][k];
      w2[r][k] = acc;
    }
    __syncthreads();
    if (tid < BTILE) {
      float m = w2[tid][0];
      for (int k = 1; k < Kn; ++k) m = fmaxf(m, w2[tid][k]);
      float s = 0.0f, ex[Kn];
      for (int k = 0; k < Kn; ++k) { ex[k] = expf(w2[tid][k] - m); s += ex[k]; }
      for (int k = 0; k < Kn; ++k) w2[tid][k] = ex[k] / s;
      // top-2 (first-occurrence tie-break like lax.top_k / argmax)
      int i0 = 0; float v0 = w2[tid][0];
      for (int k = 1; k < Kn; ++k) if (w2[tid][k] > v0) { v0 = w2[tid][k]; i0 = k; }
      int i1 = (i0 == 0) ? 1 : 0; float v1 = w2[tid][i1];
      for (int k = 0; k < Kn; ++k)
        if (k != i0 && w2[tid][k] > v1) { v1 = w2[tid][k]; i1 = k; }
      float denom = fmaxf(v0 + v1, EPS);
      idx_l[tid][0] = i0; idx_l[tid][1] = i1;
      wn_l[tid][0] = v0 / denom; wn_l[tid][1] = v1 / denom;
    }
    __syncthreads();

    // ---- decode: logits[b,v] = sum_{c2} wn * (state . rw[k,v,:] + rb[k,v]) ----
    if (tid < BTILE * Sn) {
      int r = tid >> 2, v = tid & 3;
      float out = 0.0f;
      for (int c2 = 0; c2 < 2; ++c2) {
        int k = idx_l[r][c2];
        const float* rwp = rw + ((size_t)(k * Sn + v)) * Dn;
        float acc = rb[k * Sn + v];
        for (int d = 0; d < DQn; ++d) acc += q[r][d] * rwp[d];
        for (int d = 0; d < DQn; ++d) acc += p[r][d] * rwp[DQn + d];
        acc += s_col[r] * rwp[2 * DQn];
        out += wn_l[r][c2] * acc;
      }
      logits4[r][v] = out;
    }
    __syncthreads();

    // ---- per-row losses & accumulation ----
    if (tid < BTILE) {
      int r = tid;
      int task = task_t[r], target = tgt_t[r], prev = prev_t[r];
      float l0 = logits4[r][0], l1 = logits4[r][1];
      float l2 = logits4[r][2], l3 = logits4[r][3];
      // ce2 (JAX clamps OOB gather index -> min(target,1))
      float m2 = fmaxf(l0, l1);
      float lse2 = m2 + logf(expf(l0 - m2) + expf(l1 - m2));
      int t2 = (target < 1) ? target : 1;
      float ce2 = lse2 - ((t2 == 0) ? l0 : l1);
      // ce4
      float m4 = fmaxf(fmaxf(l0, l1), fmaxf(l2, l3));
      float lse4 = m4 + logf(expf(l0 - m4) + expf(l1 - m4) +
                             expf(l2 - m4) + expf(l3 - m4));
      float lt = (target == 0) ? l0 : (target == 1) ? l1 : (target == 2) ? l2 : l3;
      float ce4 = lse4 - lt;
      float pred_error = (task == 0) ? ce2 : ce4;
      int pred;
      if (task == 0) { pred = (l1 > l0) ? 1 : 0; }
      else {
        pred = 0; float pm = l0;
        if (l1 > pm) { pm = l1; pred = 1; }
        if (l2 > pm) { pm = l2; pred = 2; }
        if (l3 > pm) { pm = l3; pred = 3; }
      }
      float correct = (pred == target) ? 1.0f : 0.0f;
      float csl = -logf(fmaxf(wlast[r][prev], EPS));
      float F = GAMMA * (pred_error + csl);
      atomicAdd(&accum[0], F);
      atomicAdd(&accum[1], correct);
      atomicAdd(&accum[2], pred_error);
      for (int k = 0; k < Kn; ++k) atomicAdd(&accum[3 + k], wlast[r][k]);
      atomicAdd(&accum[11 + task], correct);
      atomicAdd(&accum[13 + task], 1.0f);
      if (task == 1) {
        for (int k = 0; k < Kn; ++k) atomicAdd(&accum[15 + prev * Kn + k], wlast[r][k]);
        atomicAdd(&accum[47 + prev], 1.0f);
      }
    }
    __syncthreads();
  } // t

  if (tid < NACC) atomicAdd(&ws_accum[tid], accum[tid]);
}

__global__ void seqmodel_finalize(const float* __restrict__ ws, float* __restrict__ out) {
  if (threadIdx.x != 0 || blockIdx.x != 0) return;
  const float EPS = 1e-8f;
  const float inv = 1.0f / (float)(Bn * Tn);
  out[0] = ws[0] * inv;            // mean_F
  out[1] = ws[1] * inv;            // mean_acc
  out[2] = ws[2] * inv;            // mean_ce
  float usum = 0.0f;
  for (int k = 0; k < Kn; ++k) { out[3 + k] = ws[3 + k]; usum += ws[3 + k]; }
  float ent = 0.0f;
  for (int k = 0; k < Kn; ++k) {
    float d = ws[3 + k] / (usum + EPS);
    ent -= d * logf(d + EPS);
  }
  out[11] = ent;                   // chart_ent
  out[12] = ws[11] / (ws[13] + EPS);   // per_task_acc[0]
  out[13] = ws[12] / (ws[14] + EPS);   // per_task_acc[1]
  float ssum = 0.0f;
  for (int i = 0; i < Sn * Kn; ++i) { out[14 + i] = ws[15 + i]; ssum += ws[15 + i]; }
  for (int s = 0; s < Sn; ++s) out[46 + s] = ws[47 + s];   // scnt
  float ps[Sn], pc[Kn], joint[Sn][Kn];
  for (int s = 0; s < Sn; ++s) ps[s] = 0.0f;
  for (int k = 0; k < Kn; ++k) pc[k] = 0.0f;
  for (int s = 0; s < Sn; ++s)
    for (int k = 0; k < Kn; ++k) {
      joint[s][k] = ws[15 + s * Kn + k] / (ssum + EPS);
      ps[s] += joint[s][k]; pc[k] += joint[s][k];
    }
  float mi = 0.0f;
  for (int s = 0; s < Sn; ++s)
    for (int k = 0; k < Kn; ++k)
      mi += joint[s][k] * (logf(joint[s][k] + EPS) - logf(ps[s] + EPS) - logf(pc[k] + EPS));
  out[50] = (ssum > 0.0f) ? fmaxf(mi, 0.0f) : 0.0f;  // mi
}

extern "C" void kernel_launch(void* const* d_in, const int* in_sizes, int n_in,
                              void* d_out, int out_size, void* d_ws, size_t ws_size,
                              hipStream_t stream) {
  (void)in_sizes; (void)n_in; (void)out_size; (void)ws_size;
  const int*   inputs   = (const int*)d_in[0];
  const int*   targets  = (const int*)d_in[1];
  const int*   task_ids = (const int*)d_in[2];
  const float* init_s   = (const float*)d_in[3];
  const float* bit_e    = (const float*)d_in[4];
  const float* task_e   = (const float*)d_in[5];
  const float* rce      = (const float*)d_in[6];
  const float* rw       = (const float*)d_in[7];
  const float* rb       = (const float*)d_in[8];
  const float* Wq       = (const float*)d_in[9];
  const float* Wc       = (const float*)d_in[10];
  const float* rbias    = (const float*)d_in[11];
  const float* A        = (const float*)d_in[12];
  float* ws  = (float*)d_ws;
  float* out = (float*)d_out;

  seqmodel_init<<<1, 64, 0, stream>>>(ws);
  seqmodel_main<<<NWG, 256, 0, stream>>>(inputs, targets, task_ids, init_s, bit_e, task_e,
                                         rce, rw, rb, Wq, Wc, rbias, A, ws);
  seqmodel_finalize<<<1, 64, 0, stream>>>(ws, out);
}